// PEER_20882130993294
// MI455X (gfx1250) — compile-verified
//
#include <hip/hip_runtime.h>
#include <math.h>

typedef float v2f __attribute__((ext_vector_type(2)));
typedef float v8f __attribute__((ext_vector_type(8)));

#define HEADS 8
#define DIM 1024
#define DIM_KEY 512
#define NUM_KEYS 256
#define TOPK 16
#define SROW 260          // padded LDS row (260 % 64 = 4 -> bank stagger)
#define NEG_INF (-3.4e38f)

// ---------------------------------------------------------------------------
// Kernel 1: K2[r][e] = sum_d keys[h][k][p][d] * Wq[(p*8+h)*512 + d][e]
//           r = (p*8 + h)*256 + k   (4096 rows x 1024 cols)
// ---------------------------------------------------------------------------
__global__ __launch_bounds__(256) void k2_build(const float* __restrict__ Wq,
                                                const float* __restrict__ keys,
                                                float* __restrict__ K2) {
  const int r = blockIdx.x;
  const int p = r >> 11;
  const int h = (r >> 8) & 7;
  const int k = r & 255;
  const float* krow  = keys + (((size_t)h * NUM_KEYS + k) * 2 + p) * DIM_KEY;
  const float* wbase = Wq + (size_t)(p * HEADS + h) * DIM_KEY * DIM;
  __shared__ float ks[DIM_KEY];
  for (int i = threadIdx.x; i < DIM_KEY; i += 256) ks[i] = krow[i];
  __syncthreads();
  const int e = threadIdx.x;
  float a0 = 0.f, a1 = 0.f, a2 = 0.f, a3 = 0.f;
  for (int d = 0; d < DIM_KEY; ++d) {
    const float kv = ks[d];
    const float* w = wbase + (size_t)d * DIM;
    a0 += kv * w[e];
    a1 += kv * w[e + 256];
    a2 += kv * w[e + 512];
    a3 += kv * w[e + 768];
  }
  float* o = K2 + (size_t)r * DIM;
  o[e] = a0; o[e + 256] = a1; o[e + 512] = a2; o[e + 768] = a3;
}

// ---------------------------------------------------------------------------
// Wave helpers for top-k
// ---------------------------------------------------------------------------
__device__ __forceinline__ float waveArgMax(float m, int mi, int* outIdx) {
  #pragma unroll
  for (int off = 16; off > 0; off >>= 1) {
    float om = __shfl_xor(m, off, 32);
    int   oi = __shfl_xor(mi, off, 32);
    if (om > m || (om == m && oi < mi)) { m = om; mi = oi; }
  }
  *outIdx = mi;
  return m;
}

// top-16 (descending) of 256 LDS floats; lane0 writes results to dst[0..15]
__device__ __forceinline__ void topk16_lds(const float* __restrict__ srow,
                                           float* __restrict__ dst, int lane) {
  float v[8];
  #pragma unroll
  for (int j = 0; j < 8; ++j) v[j] = srow[lane + 32 * j];
  #pragma unroll
  for (int sel = 0; sel < 16; ++sel) {
    float m = NEG_INF; int mi = 0x7fffffff;
    #pragma unroll
    for (int j = 0; j < 8; ++j) {
      const int idx = lane + 32 * j;
      if (v[j] > m) { m = v[j]; mi = idx; }
    }
    int gi; float gm = waveArgMax(m, mi, &gi);
    #pragma unroll
    for (int j = 0; j < 8; ++j)
      if (gi == lane + 32 * j) v[j] = NEG_INF;
    if (lane == 0) dst[sel] = gm;
  }
}

// combined top-16 over the 16x16 outer-sum grid, softmax, store weights+indices
__device__ __forceinline__ void topk16_combined(const float* __restrict__ sx,
                                                const float* __restrict__ sy,
                                                int lane, float* __restrict__ outw,
                                                int* __restrict__ outi,
                                                int token, int h) {
  float v[8];
  #pragma unroll
  for (int j = 0; j < 8; ++j) {
    const int c = lane + 32 * j;         // c = i*16 + j  (i = c>>4, j = c&15)
    v[j] = sx[c >> 4] + sy[c & 15];
  }
  float sc[16]; int ci[16];
  #pragma unroll
  for (int sel = 0; sel < 16; ++sel) {
    float m = NEG_INF; int mi = 0x7fffffff;
    #pragma unroll
    for (int j = 0; j < 8; ++j) {
      const int c = lane + 32 * j;
      if (v[j] > m) { m = v[j]; mi = c; }
    }
    int gi; float gm = waveArgMax(m, mi, &gi);
    #pragma unroll
    for (int j = 0; j < 8; ++j)
      if (gi == lane + 32 * j) v[j] = NEG_INF;
    sc[sel] = gm; ci[sel] = gi;
  }
  float s0 = sc[0], wsum = 0.f, w[16];
  #pragma unroll
  for (int t = 0; t < 16; ++t) { w[t] = __expf(sc[t] - s0); wsum += w[t]; }
  const float inv = 1.f / wsum;
  if (lane == 0) {
    const size_t base = ((size_t)token * HEADS + h) * TOPK;
    #pragma unroll
    for (int t = 0; t < 16; ++t) { outw[base + t] = w[t] * inv; outi[base + t] = ci[t]; }
  }
}

// ---------------------------------------------------------------------------
// Kernel 2: fused sim-GEMM (f32 WMMA) + dual top-k + combine + softmax.
// Block = 256 threads = 8 waves, owns 16 tokens. Per head: S[16x512] in LDS.
// Waves 0-3 cover the p=0 half (cols 0..255), waves 4-7 the p=1 half.
// ---------------------------------------------------------------------------
__global__ __launch_bounds__(256) void sim_topk(const float* __restrict__ X,
                                                const float* __restrict__ K2,
                                                float* __restrict__ wts,
                                                int* __restrict__ pidx) {
  __shared__ float s_s[2][16][SROW];
  __shared__ float s_top[8][2][16];
  const int tid  = threadIdx.x;
  const int lane = tid & 31;
  const int wave = tid >> 5;
  const int lr   = lane & 15;
  const int hi   = lane >> 4;
  const int t0   = blockIdx.x * 16;
  const int p    = wave >> 2;            // column half owned by this wave
  const int lc0  = (wave & 3) * 64;      // col offset inside the half

  const float* arow = X + (size_t)(t0 + lr) * DIM + 2 * hi;

  for (int h = 0; h < HEADS; ++h) {
    // ---- GEMM: this wave computes S rows 0..15, cols [p*256+lc0, +64) ----
    v8f c0 = {}, c1 = {}, c2 = {}, c3 = {};
    const size_t rb = ((size_t)(p * HEADS + h) * NUM_KEYS + lc0 + lr) * DIM + 2 * hi;
    const float* b0 = K2 + rb;
    const float* b1 = K2 + rb + (size_t)16 * DIM;
    const float* b2 = K2 + rb + (size_t)32 * DIM;
    const float* b3 = K2 + rb + (size_t)48 * DIM;
    for (int k0 = 0; k0 < DIM; k0 += 4) {
      v2f a   = *(const v2f*)(arow + k0);
      v2f vb0 = *(const v2f*)(b0 + k0);
      v2f vb1 = *(const v2f*)(b1 + k0);
      v2f vb2 = *(const v2f*)(b2 + k0);
      v2f vb3 = *(const v2f*)(b3 + k0);
      c0 = __builtin_amdgcn_wmma_f32_16x16x4_f32(false, a, false, vb0, (short)0, c0, false, false);
      c1 = __builtin_amdgcn_wmma_f32_16x16x4_f32(false, a, false, vb1, (short)0, c1, false, false);
      c2 = __builtin_amdgcn_wmma_f32_16x16x4_f32(false, a, false, vb2, (short)0, c2, false, false);
      c3 = __builtin_amdgcn_wmma_f32_16x16x4_f32(false, a, false, vb3, (short)0, c3, false, false);
    }
    // D-frag layout: VGPR v -> row = v + 8*hi, col = lr
    #pragma unroll
    for (int v = 0; v < 8; ++v) {
      const int row = v + 8 * hi;
      s_s[p][row][lc0 +  0 + lr] = c0[v];
      s_s[p][row][lc0 + 16 + lr] = c1[v];
      s_s[p][row][lc0 + 32 + lr] = c2[v];
      s_s[p][row][lc0 + 48 + lr] = c3[v];
    }
    __syncthreads();

    // ---- top-k: wave handles rows 2*wave and 2*wave+1 ----
    for (int rr = 0; rr < 2; ++rr) {
      const int row = wave * 2 + rr;
      topk16_lds(&s_s[0][row][0], &s_top[wave][0][0], lane);
      topk16_lds(&s_s[1][row][0], &s_top[wave][1][0], lane);
      __syncthreads();
      topk16_combined(&s_top[wave][0][0], &s_top[wave][1][0], lane,
                      wts, pidx, t0 + row, h);
      __syncthreads();
    }
  }
}

// ---------------------------------------------------------------------------
// Kernel 3: per-token expert stage. h = gelu(x . W_down[idx]) * weight;
//           out = sum over 128 (head,slot) entries of h * W_up[idx].
// ---------------------------------------------------------------------------
__global__ __launch_bounds__(256) void expert_out(const float* __restrict__ X,
                                                  const float* __restrict__ Wd,
                                                  const float* __restrict__ Wu,
                                                  const float* __restrict__ wts,
                                                  const int* __restrict__ pidx,
                                                  float* __restrict__ out) {
  __shared__ float xs[DIM];
  __shared__ float coeff[HEADS * TOPK];
  __shared__ int   eidx[HEADS * TOPK];
  const int tid = threadIdx.x, lane = tid & 31, wave = tid >> 5;
  const size_t token = blockIdx.x;
  const float* xr = X + token * DIM;
  #pragma unroll
  for (int j = 0; j < 4; ++j) xs[tid + 256 * j] = xr[tid + 256 * j];
  if (tid < HEADS * TOPK) {
    const size_t base = token * HEADS * TOPK;
    eidx[tid]  = pidx[base + tid];
    coeff[tid] = wts[base + tid];          // softmax weight; gelu folded in below
  }
  __syncthreads();

  for (int e = wave * 16; e < wave * 16 + 16; ++e) {
    const float* wd = Wd + (size_t)eidx[e] * DIM;
    float s = 0.f;
    #pragma unroll
    for (int j = 0; j < 32; ++j) s += xs[lane + 32 * j] * wd[lane + 32 * j];
    #pragma unroll
    for (int off = 16; off > 0; off >>= 1) s += __shfl_xor(s, off, 32);
    if (lane == 0) {
      const float g = 0.5f * s * (1.f + erff(s * 0.70710678118654752f));  // exact gelu
      coeff[e] *= g;
    }
  }
  __syncthreads();

  float a0 = 0.f, a1 = 0.f, a2 = 0.f, a3 = 0.f;
  for (int e = 0; e < HEADS * TOPK; ++e) {
    const float c = coeff[e];
    const float* wu = Wu + (size_t)eidx[e] * DIM;
    a0 += c * wu[tid];
    a1 += c * wu[tid + 256];
    a2 += c * wu[tid + 512];
    a3 += c * wu[tid + 768];
  }
  float* o = out + token * DIM;
  o[tid] = a0; o[tid + 256] = a1; o[tid + 512] = a2; o[tid + 768] = a3;
}

// ---------------------------------------------------------------------------
extern "C" void kernel_launch(void* const* d_in, const int* in_sizes, int n_in,
                              void* d_out, int out_size, void* d_ws, size_t ws_size,
                              hipStream_t stream) {
  const float* x    = (const float*)d_in[0];
  const float* Wq   = (const float*)d_in[1];
  const float* keys = (const float*)d_in[2];
  const float* Wd   = (const float*)d_in[3];
  const float* Wu   = (const float*)d_in[4];
  float* out = (float*)d_out;

  const size_t ntok = (size_t)in_sizes[0] / DIM;        // b*n = 8192
  char* ws = (char*)d_ws;
  float* K2 = (float*)ws;                               // 4096*1024 f32 = 16 MB
  const size_t k2_bytes = (size_t)2 * HEADS * NUM_KEYS * DIM * sizeof(float);
  float* wts = (float*)(ws + k2_bytes);                 // ntok*8*16 f32 = 4 MB
  const size_t w_bytes = ntok * HEADS * TOPK * sizeof(float);
  int* pidx = (int*)(ws + k2_bytes + w_bytes);          // ntok*8*16 i32 = 4 MB

  k2_build<<<2 * HEADS * NUM_KEYS, 256, 0, stream>>>(Wq, keys, K2);
  sim_topk<<<(int)(ntok / 16), 256, 0, stream>>>(x, K2, wts, pidx);
  expert_out<<<(int)ntok, 256, 0, stream>>>(x, Wd, Wu, wts, pidx, out);
}